// StepwiseVectorizedConstantVelocityModel_40587440947785
// MI455X (gfx1250) — compile-verified
//
#include <hip/hip_runtime.h>
#include <math.h>

#define NN 512
#define SS 64
#define MM 32768
#define DELTA_F 1.5625f       /* 100/64, exact in binary */
#define EPS_F 1e-12f
#define SQRTPI_F 1.7724538509055160273f

typedef float v2f __attribute__((ext_vector_type(2)));
typedef float v8f __attribute__((ext_vector_type(8)));

/* ---- stats float-array layout (base = fb + 65536), 64 entries each ----
   [0]C0 [64]C1 [128]C2 [192]A0 [256]A1 [320]B0 [384]B1
   [448]Sdd [512]Sdd2 [576]ts [640]T                                      */

/* Kernel 1: Z_steps prefix sums, layout Zs[s][d][n] (n-contiguous). */
__global__ void k_prefix(const float* __restrict__ z0,
                         const float* __restrict__ v0,
                         float* __restrict__ Zs) {
    int n = blockIdx.x * blockDim.x + threadIdx.x;
    if (n >= NN) return;
    float zd0 = z0[n * 2 + 0];
    float zd1 = z0[n * 2 + 1];
    for (int s = 0; s < SS; ++s) {
        Zs[(s * 2 + 0) * NN + n] = zd0;
        Zs[(s * 2 + 1) * NN + n] = zd1;
        float vd0 = v0[n * 2 * SS + 0 * SS + s];
        float vd1 = v0[n * 2 * SS + 1 * SS + s];
        zd0 += vd0 * DELTA_F;
        zd1 += vd1 * DELTA_F;
    }
}

/* Kernel 2: per-step coefficients for the closed-form event term.
   C0=Σ|z|², C1=Σ z·v, C2=Σ|v|², A_d=Σ z_d, B_d=Σ v_d (sums over n). */
__global__ void k_stepstats(const float* __restrict__ Zs,
                            const float* __restrict__ v0,
                            float* __restrict__ stats) {
    int s = blockIdx.x, tid = threadIdx.x;
    float c0 = 0.f, c1 = 0.f, c2 = 0.f, a0 = 0.f, a1 = 0.f, b0 = 0.f, b1 = 0.f;
    for (int n = tid; n < NN; n += 256) {
        float z0v = Zs[(s * 2 + 0) * NN + n];
        float z1v = Zs[(s * 2 + 1) * NN + n];
        float w0 = v0[n * 2 * SS + s];
        float w1 = v0[n * 2 * SS + SS + s];
        c0 += z0v * z0v + z1v * z1v;
        c1 += z0v * w0 + z1v * w1;
        c2 += w0 * w0 + w1 * w1;
        a0 += z0v; a1 += z1v; b0 += w0; b1 += w1;
    }
    __shared__ float red[7][256];
    red[0][tid] = c0; red[1][tid] = c1; red[2][tid] = c2;
    red[3][tid] = a0; red[4][tid] = a1; red[5][tid] = b0; red[6][tid] = b1;
    __syncthreads();
    for (int st = 128; st > 0; st >>= 1) {
        if (tid < st)
            for (int k = 0; k < 7; ++k) red[k][tid] += red[k][tid + st];
        __syncthreads();
    }
    if (tid == 0)
        for (int k = 0; k < 7; ++k) stats[k * 64 + s] = red[k][0];
}

/* Kernel 3: per-step event statistics: count, Σdd, Σdd², first (=min) time. */
__global__ void k_events(const float* __restrict__ data,
                         float* __restrict__ stats,
                         int* __restrict__ cntI) {
    int s = blockIdx.x, tid = threadIdx.x;
    int cnt = 0;
    float sdd = 0.f, sdd2 = 0.f, tmin = 3.4e38f;
    for (int m = tid; m < MM; m += 256) {
        float t = data[m * 3 + 2];
        float fs = floorf(t / DELTA_F);
        int idx = (int)fs;
        idx = idx < 0 ? 0 : (idx > SS - 1 ? SS - 1 : idx);
        if (idx == s) {
            float dd = t - fs * DELTA_F;
            cnt += 1; sdd += dd; sdd2 += dd * dd; tmin = fminf(tmin, t);
        }
    }
    __shared__ int ri[256];
    __shared__ float r1[256], r2[256], r3[256];
    ri[tid] = cnt; r1[tid] = sdd; r2[tid] = sdd2; r3[tid] = tmin;
    __syncthreads();
    for (int st = 128; st > 0; st >>= 1) {
        if (tid < st) {
            ri[tid] += ri[tid + st];
            r1[tid] += r1[tid + st];
            r2[tid] += r2[tid + st];
            r3[tid] = fminf(r3[tid], r3[tid + st]);
        }
        __syncthreads();
    }
    if (tid == 0) {
        cntI[s] = ri[0];
        stats[448 + s] = r1[0];
        stats[512 + s] = r2[0];
        stats[576 + s] = r3[0];
    }
}

/* Kernel 4: reverse scan (tf), T[s], occ[s], and the closed-form event term:
   event_intensity = P*b*M − N*trace + Stot  (double accumulation). */
__global__ void k_scalar(const float* __restrict__ data,
                         const float* __restrict__ beta,
                         float* __restrict__ stats,
                         const int* __restrict__ cntI,
                         int* __restrict__ occI,
                         double* __restrict__ evInt) {
    float carry = data[(MM - 1) * 3 + 2];
    for (int s = SS - 1; s >= 0; --s) {
        int occ = cntI[s] > 0;
        occI[s] = occ;
        float ts = stats[576 + s];
        stats[640 + s] = carry - ts;   /* T[s]; only used where occ */
        if (occ) carry = ts;
    }
    double b = (double)beta[0];
    double trace = 0.0, stot = 0.0;
    for (int s = 0; s < SS; ++s) {
        double c  = (double)cntI[s];
        double sd = (double)stats[448 + s];
        double s2 = (double)stats[512 + s];
        double C0 = stats[0 + s],  C1 = stats[64 + s],  C2 = stats[128 + s];
        double A0 = stats[192 + s], A1 = stats[256 + s];
        double B0 = stats[320 + s], B1 = stats[384 + s];
        trace += c * C0 + 2.0 * sd * C1 + s2 * C2;
        stot  += c * (A0 * A0 + A1 * A1) + 2.0 * sd * (A0 * B0 + A1 * B1)
               + s2 * (B0 * B0 + B1 * B1);
    }
    double P = (double)NN * (NN - 1) / 2.0;
    evInt[0] = P * b * (double)MM - (double)NN * trace + stot;
}

/* Kernel 5: non-event term. One wave per 16x16 pair tile; per step s, four
   V_WMMA_F32_16X16X4_F32 Gram products (zz, zv, vz, vv) with K=2 used. */
__global__ void __launch_bounds__(32)
k_tiles(const float* __restrict__ Zs,
        const float* __restrict__ v0,
        const float* __restrict__ beta,
        const float* __restrict__ Tarr,
        const int* __restrict__ occI,
        double* __restrict__ tileSum) {
    int bn = blockIdx.x, bm = blockIdx.y;
    double* out = &tileSum[bn * 32 + bm];
    int lane = threadIdx.x;
    if (bm < bn) { if (lane == 0) *out = 0.0; return; }
    int n0 = bn * 16, m0 = bm * 16;

    /* Row stats in LDS, padded stride 65 -> conflict-free strided reads. */
    __shared__ float lds[6][16 * 65];
    for (int i = lane; i < 1024; i += 32) {
        int s = i >> 4, row = i & 15;
        float z0v = Zs[(s * 2 + 0) * NN + n0 + row];
        float z1v = Zs[(s * 2 + 1) * NN + n0 + row];
        float w0  = v0[(n0 + row) * 2 * SS + s];
        float w1  = v0[(n0 + row) * 2 * SS + SS + s];
        lds[0][row * 65 + s] = z0v * z0v + z1v * z1v;
        lds[1][row * 65 + s] = z0v * w0 + z1v * w1;
        lds[2][row * 65 + s] = w0 * w0 + w1 * w1;
        z0v = Zs[(s * 2 + 0) * NN + m0 + row];
        z1v = Zs[(s * 2 + 1) * NN + m0 + row];
        w0  = v0[(m0 + row) * 2 * SS + s];
        w1  = v0[(m0 + row) * 2 * SS + SS + s];
        lds[3][row * 65 + s] = z0v * z0v + z1v * z1v;
        lds[4][row * 65 + s] = z0v * w0 + z1v * w1;
        lds[5][row * 65 + s] = w0 * w0 + w1 * w1;
    }
    __syncthreads();
    __builtin_prefetch(Zs, 0, 1);

    float bval = beta[0];
    int lm = lane & 15;
    int half = lane >> 4;
    float acc = 0.f;

    for (int s = 0; s < SS; ++s) {
        if (!occI[s]) continue;
        float Tv = Tarr[s];
        /* A/B fragments: lanes 0-15 carry K=0,1 (d components); lanes 16-31
           carry K=2,3 = 0 (zero-padded K dimension). */
        float a0 = Zs[(s * 2 + 0) * NN + n0 + lm];
        float a1 = Zs[(s * 2 + 1) * NN + n0 + lm];
        float w0 = v0[(n0 + lm) * 2 * SS + s];
        float w1 = v0[(n0 + lm) * 2 * SS + SS + s];
        float c0 = Zs[(s * 2 + 0) * NN + m0 + lm];
        float c1 = Zs[(s * 2 + 1) * NN + m0 + lm];
        float u0 = v0[(m0 + lm) * 2 * SS + s];
        float u1 = v0[(m0 + lm) * 2 * SS + SS + s];
        bool lo = (half == 0);
        v2f az, av, bz, bv;
        az.x = lo ? a0 : 0.f;  az.y = lo ? a1 : 0.f;
        av.x = lo ? w0 : 0.f;  av.y = lo ? w1 : 0.f;
        bz.x = lo ? c0 : 0.f;  bz.y = lo ? c1 : 0.f;
        bv.x = lo ? u0 : 0.f;  bv.y = lo ? u1 : 0.f;
        v8f zc = {0.f, 0.f, 0.f, 0.f, 0.f, 0.f, 0.f, 0.f};
        v8f czz = __builtin_amdgcn_wmma_f32_16x16x4_f32(false, az, false, bz, (short)0, zc, false, false);
        v8f czv = __builtin_amdgcn_wmma_f32_16x16x4_f32(false, az, false, bv, (short)0, zc, false, false);
        v8f cvz = __builtin_amdgcn_wmma_f32_16x16x4_f32(false, av, false, bz, (short)0, zc, false, false);
        v8f cvv = __builtin_amdgcn_wmma_f32_16x16x4_f32(false, av, false, bv, (short)0, zc, false, false);

        float zzm = lds[3][lm * 65 + s];
        float zvm = lds[4][lm * 65 + s];
        float vvm = lds[5][lm * 65 + s];
#pragma unroll
        for (int r = 0; r < 8; ++r) {
            int rloc = r + 8 * half;
            float zzn = lds[0][rloc * 65 + s];
            float zvn = lds[1][rloc * 65 + s];
            float vvn = lds[2][rloc * 65 + s];
            float q  = zzn + zzm - 2.f * czz[r];
            float p  = zvn + zvm - czv[r] - cvz[r];
            float rr = sqrtf(vvn + vvm - 2.f * cvv[r] + EPS_F);
            float por = p / rr;
            float term = expf(bval - q + por * por) * (0.5f * SQRTPI_F / rr)
                       * (erff(rr * Tv + por) - erff(por));
            int gn = n0 + rloc, gm = m0 + lm;
            if (gn < gm) acc += term;
        }
    }

    __shared__ double rsum[32];
    rsum[lane] = (double)acc;
    __syncthreads();
    if (lane == 0) {
        double t = 0.0;
        for (int i = 0; i < 32; ++i) t += rsum[i];
        *out = t;
    }
}

/* Kernel 6: deterministic final reduction + output. */
__global__ void k_final(const double* __restrict__ tileSum,
                        const double* __restrict__ evInt,
                        float* __restrict__ out) {
    __shared__ double sh[256];
    int tid = threadIdx.x;
    double t = 0.0;
    for (int i = tid; i < 1024; i += 256) t += tileSum[i];
    sh[tid] = t;
    __syncthreads();
    for (int st = 128; st > 0; st >>= 1) {
        if (tid < st) sh[tid] += sh[tid + st];
        __syncthreads();
    }
    if (tid == 0) out[0] = (float)(evInt[0] - sh[0]);
}

extern "C" void kernel_launch(void* const* d_in, const int* in_sizes, int n_in,
                              void* d_out, int out_size, void* d_ws, size_t ws_size,
                              hipStream_t stream) {
    const float* data = (const float*)d_in[0];
    /* d_in[1]=t0, d_in[2]=tn unused by the forward computation */
    const float* z0   = (const float*)d_in[3];
    const float* v0   = (const float*)d_in[4];
    const float* beta = (const float*)d_in[5];
    float* out = (float*)d_out;

    double* wsd     = (double*)d_ws;
    double* evInt   = wsd;          /* 1 double  */
    double* tileSum = wsd + 8;      /* 1024 doubles */
    float*  fb      = (float*)d_ws + 2112;  /* byte offset 8448 */
    float*  Zs      = fb;           /* [S][2][N] = 65536 floats */
    float*  stats   = fb + 65536;   /* 704 floats (see layout) */
    int*    cntI    = (int*)(fb + 65536 + 704); /* 64 ints */
    int*    occI    = cntI + 64;                /* 64 ints */

    k_prefix<<<dim3(2), dim3(256), 0, stream>>>(z0, v0, Zs);
    k_stepstats<<<dim3(SS), dim3(256), 0, stream>>>(Zs, v0, stats);
    k_events<<<dim3(SS), dim3(256), 0, stream>>>(data, stats, cntI);
    k_scalar<<<dim3(1), dim3(1), 0, stream>>>(data, beta, stats, cntI, occI, evInt);
    k_tiles<<<dim3(32, 32), dim3(32), 0, stream>>>(Zs, v0, beta, stats + 640, occI, tileSum);
    k_final<<<dim3(1), dim3(256), 0, stream>>>(tileSum, evInt, out);
}